// SpiralTwistedInteractionHead_27393301413952
// MI455X (gfx1250) — compile-verified
//
#include <hip/hip_runtime.h>
#include <hip/hip_bf16.h>
#include <math.h>

// ---------------------------------------------------------------------------
// SpiralTwistedInteractionHead for MI455X (gfx1250, wave32, WMMA)
//
// Pass 1: 1x1-conv GEMM (M=100352, N=8->pad16, K=256) via V_WMMA_F32_16X16X4_F32
//         fused BN(inference)+ReLU epilogue, coalesced store of z to workspace.
//         2 tiles (32 px) per wave + K-unroll 4 => ~6-7 MB of loads in flight
//         across 3136 waves to saturate 23.3 TB/s HBM.
// Pass 2: depthwise 3x3 * scale, channel L2 norms, pairwise interactions,
//         44-channel output.
// ---------------------------------------------------------------------------

#define C_RED   8
#define CIN     256
#define HH      56
#define WW      56
#define HW      (HH * WW)          // 3136
#define NBATCH  32
#define NPIX    (NBATCH * HW)      // 100352 total pixels (GEMM M)
#define BN_EPS  1e-5f
#define NORM_EPS 1e-6f

typedef __attribute__((ext_vector_type(2))) float v2f;
typedef __attribute__((ext_vector_type(8))) float v8f;

// Padded per-pixel stride in the LDS transpose buffer (9 floats): gcd(9,64)=1
// makes both the half-wave writers and the read-back conflict-free.
#define ZS_STRIDE 9

// ---------------------------------------------------------------------------
// Kernel 1: WMMA GEMM + BN + ReLU  ->  z [32][8][3136] fp32 in workspace
//   block = 256 threads (8 waves). Each wave owns 32 consecutive pixels
//   (2 WMMA tiles of 16). 32 | 3136, so a wave never straddles a batch image.
// ---------------------------------------------------------------------------
__global__ __launch_bounds__(256) void sth_gemm_bn_relu_kernel(
    const float* __restrict__ x,        // [32][256][3136]
    const float* __restrict__ w,        // [8][256]
    const float* __restrict__ bn_gamma, // [8]
    const float* __restrict__ bn_beta,  // [8]
    const float* __restrict__ bn_mean,  // [8]
    const float* __restrict__ bn_var,   // [8]
    float* __restrict__ z)              // [32][8][3136]
{
    __shared__ float Wlds[16 * CIN];                 // zero-padded weights, 16 KB
    __shared__ float zstage[8][32 * ZS_STRIDE];      // per-wave transpose staging

    const int tid = threadIdx.x;

    // Cooperative load of weights into LDS, channels 8..15 zero-padded.
    for (int i = tid; i < 16 * CIN; i += 256) {
        const int n = i >> 8;          // padded output channel
        const int k = i & (CIN - 1);   // input channel
        Wlds[i] = (n < C_RED) ? w[n * CIN + k] : 0.0f;
    }
    __syncthreads();

    const int wave = tid >> 5;
    const int lane = tid & 31;
    const int n    = lane & 15;        // D-matrix column = output channel
    const int hi   = lane >> 4;        // half-wave selector (K sub-pair)

    const long gp0  = ((long)blockIdx.x * 8 + wave) * 32;  // first pixel of chunk
    const int  b    = (int)(gp0 / HW);
    const int  pix0 = (int)(gp0 % HW);
    const float* xb = x + (size_t)b * CIN * HW + pix0 + n; // n==lane&15 is also A-row M

    // Per-lane BN affine constants for channel n (zero for padded channels).
    float bnInv = 0.0f, bnBias = 0.0f;
    if (n < C_RED) {
        bnInv  = bn_gamma[n] * rsqrtf(bn_var[n] + BN_EPS);
        bnBias = bn_beta[n] - bn_mean[n] * bnInv;
    }

    v8f c0 = {}, c1 = {};

    const float* wrow = &Wlds[n * CIN];

#pragma unroll 4
    for (int k = 0; k < CIN; k += 4) {
        const int ks = k + 2 * hi;     // this half-wave's K pair: ks, ks+1
        // B fragment: lanes 0-15 -> K=k+0/k+1 ; lanes 16-31 -> K=k+2/k+3
        const v2f bfrag = *(const v2f*)(wrow + ks);
        // A fragments for the two 16-pixel tiles (same layout as B, M=lane&15)
        const float* ap = xb + (size_t)ks * HW;
        v2f a0, a1;
        a0[0] = ap[0];   a0[1] = ap[HW + 0];
        a1[0] = ap[16];  a1[1] = ap[HW + 16];

        c0 = __builtin_amdgcn_wmma_f32_16x16x4_f32(false, a0, false, bfrag,
                                                   (short)0, c0, false, false);
        c1 = __builtin_amdgcn_wmma_f32_16x16x4_f32(false, a1, false, bfrag,
                                                   (short)0, c1, false, false);
    }

    // Epilogue: BN + ReLU in registers, transpose via LDS.
    // D layout: lane column N=n, VGPR r -> pixel (tile*16 + 8*hi + r).
    float* zsw = &zstage[wave][0];
    if (n < C_RED) {
#pragma unroll
        for (int r = 0; r < 8; ++r) {
            const int p0 = 0 * 16 + 8 * hi + r;
            const int p1 = 1 * 16 + 8 * hi + r;
            zsw[p0 * ZS_STRIDE + n] = fmaxf(c0[r] * bnInv + bnBias, 0.0f);
            zsw[p1 * ZS_STRIDE + n] = fmaxf(c1[r] * bnInv + bnBias, 0.0f);
        }
    }
    __syncthreads();

    // Coalesced global store of this wave's 32-pixel chunk, channel-major:
    // per channel, 32 lanes store 32 consecutive pixels (128 B contiguous).
    float* zb = z + (size_t)b * C_RED * HW + pix0;
#pragma unroll
    for (int ch = 0; ch < C_RED; ++ch) {
        zb[(size_t)ch * HW + lane] = zsw[lane * ZS_STRIDE + ch];
    }
}

// ---------------------------------------------------------------------------
// Kernel 2: depthwise 3x3 * scale, L2 norms, interactions -> out [32][44][3136]
//   One thread per pixel. z (3.2 MB) is L2-resident on MI455X (192 MB L2).
// ---------------------------------------------------------------------------
__global__ __launch_bounds__(256) void sth_twist_head_kernel(
    const float* __restrict__ z,      // [32][8][3136]
    const float* __restrict__ w_dw,   // [8][9]
    const float* __restrict__ scale,  // [8]
    float* __restrict__ out)          // [32][44][3136]
{
    const int gp = blockIdx.x * 256 + threadIdx.x;   // < NPIX
    const int b  = gp / HW;
    const int p  = gp % HW;
    const int y  = p / WW;
    const int xq = p % WW;

    const float* zb = z + (size_t)b * C_RED * HW;

    float zc[C_RED], zt[C_RED];
#pragma unroll
    for (int c = 0; c < C_RED; ++c) {
        const float* zp = zb + (size_t)c * HW;
        zc[c] = zp[p];
        const float* wd = w_dw + c * 9;
        float acc = 0.0f;
#pragma unroll
        for (int dy = -1; dy <= 1; ++dy) {
            const int yy = y + dy;
            if (yy < 0 || yy >= HH) continue;
#pragma unroll
            for (int dx = -1; dx <= 1; ++dx) {
                const int xx = xq + dx;
                if (xx < 0 || xx >= WW) continue;
                acc += wd[(dy + 1) * 3 + (dx + 1)] * zp[yy * WW + xx];
            }
        }
        zt[c] = acc * scale[c];
    }

    float s1 = 0.0f, s2 = 0.0f;
#pragma unroll
    for (int c = 0; c < C_RED; ++c) { s1 += zc[c] * zc[c]; s2 += zt[c] * zt[c]; }
    const float i1 = 1.0f / fmaxf(sqrtf(s1), NORM_EPS);
    const float i2 = 1.0f / fmaxf(sqrtf(s2), NORM_EPS);

    float zn[C_RED], tn[C_RED];
#pragma unroll
    for (int c = 0; c < C_RED; ++c) { zn[c] = zc[c] * i1; tn[c] = zt[c] * i2; }

    float* ob = out + (size_t)b * 44 * HW + p;
#pragma unroll
    for (int c = 0; c < C_RED; ++c) ob[(size_t)c * HW] = zn[c];

    int oc = C_RED;
#pragma unroll
    for (int i = 0; i < C_RED; ++i) {
#pragma unroll
        for (int j = i; j < C_RED; ++j) {
            ob[(size_t)oc * HW] = zn[i] * tn[j];
            ++oc;
        }
    }
}

// ---------------------------------------------------------------------------
// Launch
// ---------------------------------------------------------------------------
extern "C" void kernel_launch(void* const* d_in, const int* in_sizes, int n_in,
                              void* d_out, int out_size, void* d_ws, size_t ws_size,
                              hipStream_t stream) {
    const float* x        = (const float*)d_in[0];
    const float* w_reduce = (const float*)d_in[1];
    const float* bn_gamma = (const float*)d_in[2];
    const float* bn_beta  = (const float*)d_in[3];
    const float* bn_mean  = (const float*)d_in[4];
    const float* bn_var   = (const float*)d_in[5];
    const float* w_dw     = (const float*)d_in[6];
    const float* scale    = (const float*)d_in[7];
    float* out = (float*)d_out;
    float* z   = (float*)d_ws;                 // 32*8*3136 floats = 3.2 MB

    // Pass 1: 100352 pixels / (8 waves * 32 px) = 392 blocks.
    const int blocksA = NPIX / (8 * 32);
    sth_gemm_bn_relu_kernel<<<blocksA, 256, 0, stream>>>(
        x, w_reduce, bn_gamma, bn_beta, bn_mean, bn_var, z);

    // Pass 2: 100352 / 256 = 392 blocks.
    const int blocksB = NPIX / 256;
    sth_twist_head_kernel<<<blocksB, 256, 0, stream>>>(z, w_dw, scale, out);
}